// TwoStageMambaLayer_71648644432336
// MI455X (gfx1250) — compile-verified
//
#include <hip/hip_runtime.h>
#include <hip/hip_bf16.h>
#include <math.h>

#define D_MODEL 256
#define D_STATE 16
#define D_CONV  4
#define D_INNER 512
#define DT_RANK 16
#define B_SZ    4
#define NVAR    32
#define SEG     96
#define NTOK    (B_SZ * NVAR * SEG)   // 12288 tokens, same in both stages

typedef __attribute__((ext_vector_type(2))) float v2f;
typedef __attribute__((ext_vector_type(8))) float v8f;

// ---------------------------------------------------------------------------
// WMMA fp32 GEMM:  C[M,N] = act( A[M,K(LDA)] * W[N,K(LDW)]^T + bias[N] )
// All dimensions are compile-time so every fragment load is base-pointer +
// constant immediate offset (no per-iteration 64-bit address VALU).
// One wave computes a 16 x (16*NT) strip of C; per k-step the load phase
// issues 1 A-fragment + NT B-fragment loads together, then the compute phase
// issues NT V_WMMA_F32_16X16X4_F32 ops, letting the scheduler overlap the
// next step's loads with the current WMMAs.
// The grid covers C exactly (waves % 8 == 0 for all shapes used), so there
// is no divergent guard: EXEC is all-ones around every WMMA per the ISA.
// ACT: 0 = identity, 1 = softplus(x + bias)
// ---------------------------------------------------------------------------
template <int ACT, int NT, int M, int N, int K, int LDA, int LDW, int LDC>
__global__ void __launch_bounds__(256) wmma_gemm_f32(
    const float* __restrict__ A,
    const float* __restrict__ W,
    const float* __restrict__ bias,
    float* __restrict__ C)
{
    constexpr int NTILE = N / (16 * NT);
    int wave = (int)((blockIdx.x * blockDim.x + threadIdx.x) >> 5);
    int lane = threadIdx.x & 31;
    int tm   = (wave / NTILE) << 4;
    int tn   = (wave % NTILE) * (16 * NT);

    int half = lane >> 4;                       // 0: lanes 0-15, 1: lanes 16-31
    int lid  = lane & 15;
    int kofs = half << 1;                       // K pair = k0 + 2*half + {0,1}

    // A fragment base: row m = tm+lid
    const float* arow = A + (size_t)(tm + lid) * LDA + kofs;
    // B fragment base (B = W^T): col n = tn + 16*i + lid; i-offset is constexpr
    const float* wbase = W + (size_t)(tn + lid) * LDW + kofs;

    v8f acc[NT];
#pragma unroll
    for (int i = 0; i < NT; i++) acc[i] = (v8f){};

#pragma unroll 8
    for (int k0 = 0; k0 < K; k0 += 4) {
        // ---- load phase: 1 + NT b64 loads, issued together ----
        v2f a, b[NT];
        a.x = arow[k0];
        a.y = arow[k0 + 1];
#pragma unroll
        for (int i = 0; i < NT; i++) {
            b[i].x = wbase[(size_t)i * 16 * LDW + k0];
            b[i].y = wbase[(size_t)i * 16 * LDW + k0 + 1];
        }
        // ---- compute phase: NT WMMAs sharing the A fragment ----
#pragma unroll
        for (int i = 0; i < NT; i++) {
            acc[i] = __builtin_amdgcn_wmma_f32_16x16x4_f32(
                /*neg_a=*/false, a, /*neg_b=*/false, b[i],
                /*c_mod=*/(short)0, acc[i], /*reuse_a=*/false, /*reuse_b=*/false);
        }
    }

    // C tile: VGPR v -> row tm + v + 8*half, col tn + 16*i + lid
    size_t rbase = (size_t)(tm + (half << 3)) * LDC;
#pragma unroll
    for (int i = 0; i < NT; i++) {
        int   n  = tn + 16 * i + lid;
        float bb = (ACT == 1) ? bias[n] : 0.f;
#pragma unroll
        for (int v = 0; v < 8; v++) {
            float val = acc[i][v];
            if (ACT == 1) {
                float xv = val + bb;
                val = (xv > 20.f) ? xv : log1pf(__expf(xv));   // softplus
            }
            C[rbase + (size_t)v * LDC + n] = val;
        }
    }
}

// ---------------------------------------------------------------------------
// Causal depthwise conv (width 4) + SiLU over the first half of xz.
// xz rows are [xi(512) | z(512)].   xc[t,e] = silu(b[e] + sum_j xi[t-3+j,e]*w[e,j])
// ---------------------------------------------------------------------------
__global__ void conv_silu_kernel(const float* __restrict__ xz,
                                 const float* __restrict__ conv_w,
                                 const float* __restrict__ conv_b,
                                 float* __restrict__ xc,
                                 int Bn, int L)
{
    long idx   = (long)blockIdx.x * blockDim.x + threadIdx.x;
    long total = (long)Bn * L * D_INNER;
    if (idx >= total) return;
    int  e   = (int)(idx & (D_INNER - 1));
    long t   = idx >> 9;                 // seq*L + l
    int  l   = (int)(t % L);
    long seq = t / L;

    float acc = conv_b[e];
#pragma unroll
    for (int j = 0; j < D_CONV; j++) {
        int ls = l - (D_CONV - 1) + j;
        if (ls >= 0)
            acc += xz[(seq * L + ls) * (2 * D_INNER) + e] * conv_w[e * D_CONV + j];
    }
    xc[t * D_INNER + e] = acc / (1.f + __expf(-acc));      // silu
}

// ---------------------------------------------------------------------------
// Selective scan + D-skip + SiLU(z) gate.  One block (512 threads) per
// sequence; thread e owns channel e with its 16-wide state in registers.
// Writes gated y over the dt buffer (each thread reads dt[t,e] before
// writing y[t,e]; __syncthreads separates steps), so dt_in == y_out is OK.
// ---------------------------------------------------------------------------
__global__ void __launch_bounds__(D_INNER) scan_gate_kernel(
    const float* dt_in,                 // [Bn*L, 512]  (aliased with y_out)
    const float* __restrict__ xc,       // [Bn*L, 512]
    const float* __restrict__ dbl,      // [Bn*L, 48]  (B at +16, C at +32)
    const float* __restrict__ xz,       // [Bn*L, 1024] (z at +512)
    const float* __restrict__ A_log,    // [512, 16]
    const float* __restrict__ Dvec,     // [512]
    float* y_out,                       // [Bn*L, 512]
    int L)
{
    int seq = blockIdx.x;
    int e   = threadIdx.x;

    __shared__ float bc[2 * D_STATE];   // B_t then C_t

    float Av[D_STATE];
#pragma unroll
    for (int s = 0; s < D_STATE; s++)
        Av[s] = -__expf(A_log[e * D_STATE + s]);
    float Dsk = Dvec[e];

    float h[D_STATE];
#pragma unroll
    for (int s = 0; s < D_STATE; s++) h[s] = 0.f;

    long base = (long)seq * L;
    for (int l = 0; l < L; l++) {
        long t = base + l;
        if (threadIdx.x < 2 * D_STATE)
            bc[threadIdx.x] = dbl[t * (DT_RANK + 2 * D_STATE) + DT_RANK + threadIdx.x];
        __syncthreads();

        float dt = dt_in[t * D_INNER + e];
        float u  = xc[t * D_INNER + e];
        float du = dt * u;
        float y  = 0.f;
#pragma unroll
        for (int s = 0; s < D_STATE; s++) {
            float dA = __expf(dt * Av[s]);
            h[s] = dA * h[s] + du * bc[s];
            y   += h[s] * bc[D_STATE + s];
        }
        float z = xz[t * (2 * D_INNER) + D_INNER + e];
        float g = z / (1.f + __expf(-z));                  // silu(z)
        y_out[t * D_INNER + e] = (y + u * Dsk) * g;
        __syncthreads();
    }
}

// ---------------------------------------------------------------------------
// [Bb, V, S, D] -> [Bb, S, V, D]
// ---------------------------------------------------------------------------
__global__ void transpose_vs_kernel(const float* __restrict__ in,
                                    float* __restrict__ out,
                                    int Bb, int V, int S, int D)
{
    long idx   = (long)blockIdx.x * blockDim.x + threadIdx.x;
    long total = (long)Bb * V * S * D;
    if (idx >= total) return;
    int  d = (int)(idx % D);
    long r = idx / D;
    int  s = (int)(r % S); r /= S;
    int  v = (int)(r % V); r /= V;
    int  b = (int)r;
    out[(((long)b * S + s) * V + v) * D + d] = in[idx];
}

// ---------------------------------------------------------------------------
// Host-side launchers
// ---------------------------------------------------------------------------
template <int ACT, int NT, int M, int N, int K, int LDA, int LDW, int LDC>
static void launch_gemm(const float* A, const float* W, const float* bias,
                        float* C, hipStream_t s)
{
    constexpr int waves = (M / 16) * (N / (16 * NT));
    static_assert(waves % 8 == 0, "grid must be whole blocks of 8 waves");
    wmma_gemm_f32<ACT, NT, M, N, K, LDA, LDW, LDC>
        <<<waves / 8, 256, 0, s>>>(A, W, bias, C);
}

static void run_stage(const float* in, int Bn, int L, const float* const p[9],
                      float* xz, float* xc, float* dbl, float* dtb, float* out,
                      hipStream_t stream)
{
    const float* in_w    = p[0];   // [1024, 256]
    const float* conv_w  = p[1];   // [512, 4]
    const float* conv_b  = p[2];   // [512]
    const float* xproj_w = p[3];   // [48, 512]
    const float* dt_w    = p[4];   // [512, 16]
    const float* dt_b    = p[5];   // [512]
    const float* A_log   = p[6];   // [512, 16]
    const float* Dvec    = p[7];   // [512]
    const float* out_w   = p[8];   // [256, 512]

    // 1) xz = in @ in_w^T                       (N=1024, K=256)
    launch_gemm<0, 4, NTOK, 2 * D_INNER, D_MODEL, D_MODEL, D_MODEL, 2 * D_INNER>(
        in, in_w, nullptr, xz, stream);
    // 2) xc = silu(depthwise_conv(xi))
    {
        long total = (long)NTOK * D_INNER;
        conv_silu_kernel<<<(int)((total + 255) / 256), 256, 0, stream>>>(
            xz, conv_w, conv_b, xc, Bn, L);
    }
    // 3) dbl = xc @ xproj_w^T                   (N=48, K=512)
    launch_gemm<0, 3, NTOK, DT_RANK + 2 * D_STATE, D_INNER, D_INNER, D_INNER,
                DT_RANK + 2 * D_STATE>(xc, xproj_w, nullptr, dbl, stream);
    // 4) dt = softplus(dbl[:, :16] @ dt_w^T + dt_b)   (N=512, K=16)
    launch_gemm<1, 4, NTOK, D_INNER, DT_RANK, DT_RANK + 2 * D_STATE, DT_RANK,
                D_INNER>(dbl, dt_w, dt_b, dtb, stream);
    // 5) scan + skip + gate (y written in place over dtb)
    scan_gate_kernel<<<Bn, D_INNER, 0, stream>>>(dtb, xc, dbl, xz, A_log, Dvec, dtb, L);
    // 6) out = y @ out_w^T                      (N=256, K=512)
    launch_gemm<0, 4, NTOK, D_MODEL, D_INNER, D_INNER, D_INNER, D_MODEL>(
        dtb, out_w, nullptr, out, stream);
}

extern "C" void kernel_launch(void* const* d_in, const int* in_sizes, int n_in,
                              void* d_out, int out_size, void* d_ws, size_t ws_size,
                              hipStream_t stream)
{
    (void)in_sizes; (void)n_in; (void)out_size; (void)ws_size;

    const float* x = (const float*)d_in[0];
    const float* t_p[9];
    const float* d_p[9];
    for (int i = 0; i < 9; i++) {
        t_p[i] = (const float*)d_in[1 + i];
        d_p[i] = (const float*)d_in[10 + i];
    }

    // Workspace carve-up (floats); reused by both stages (~122 MB total)
    float* ws  = (float*)d_ws;
    float* xz  = ws;                                   // NTOK * 1024
    float* xc  = xz  + (size_t)NTOK * 2 * D_INNER;     // NTOK * 512
    float* dbl = xc  + (size_t)NTOK * D_INNER;         // NTOK * 48
    float* dtb = dbl + (size_t)NTOK * (DT_RANK + 2 * D_STATE);  // NTOK * 512
    float* s1o = dtb + (size_t)NTOK * D_INNER;         // NTOK * 256
    float* s2i = s1o + (size_t)NTOK * D_MODEL;         // NTOK * 256

    // Stage 1: time-wise scan, Bn = B*NVAR sequences of length SEG
    run_stage(x, B_SZ * NVAR, SEG, t_p, xz, xc, dbl, dtb, s1o, stream);

    // [B, NVAR, SEG, D] -> [B, SEG, NVAR, D]
    transpose_vs_kernel<<<NTOK, 256, 0, stream>>>(s1o, s2i, B_SZ, NVAR, SEG, D_MODEL);

    // Stage 2: dimension-wise scan, Bn = B*SEG sequences of length NVAR
    run_stage(s2i, B_SZ * SEG, NVAR, d_p, xz, xc, dbl, dtb, s1o, stream);

    // [B, SEG, NVAR, D] -> [B, NVAR, SEG, D]  (final output layout)
    transpose_vs_kernel<<<NTOK, 256, 0, stream>>>(s1o, (float*)d_out,
                                                  B_SZ, SEG, NVAR, D_MODEL);
}